// DICELossMultiClass3D_20658792693788
// MI455X (gfx1250) — compile-verified
//
#include <hip/hip_runtime.h>

typedef __attribute__((ext_vector_type(2))) float v2f;
typedef __attribute__((ext_vector_type(4))) float v4f;
typedef __attribute__((ext_vector_type(8))) float v8f;

#define EPS            1e-7f
#define NB             8
#define NC             4
#define BC             32                 // NB*NC slices
#define SPATIAL        (64 * 128 * 128)   // 1,048,576 floats per (b,c)
#define BLOCKS_PER_BC  64
#define THREADS        256
#define WAVES_PER_BLK  8
#define ELEMS_PER_WAVE 2048               // SPATIAL / (BLOCKS_PER_BC*WAVES_PER_BLK)
#define ITERS          16                 // ELEMS_PER_WAVE / (32 lanes * 4 floats)

// ---------------- zero the 96-float accumulator workspace ----------------
__global__ void dice_zero_ws(float* ws) {
    if (threadIdx.x < BC * 3) ws[threadIdx.x] = 0.0f;
}

// ---------------- main streaming reduction kernel ----------------
__global__ __launch_bounds__(THREADS)
void dice_partials(const float* __restrict__ prob,
                   const float* __restrict__ mask,
                   float* __restrict__ ws) {
    const int bc   = blockIdx.x / BLOCKS_PER_BC;   // (b,c) slice id, 0..31
    const int blk  = blockIdx.x % BLOCKS_PER_BC;   // block within slice
    const int wave = threadIdx.x >> 5;
    const int lane = threadIdx.x & 31;

    const int base = bc * SPATIAL + (blk * WAVES_PER_BLK + wave) * ELEMS_PER_WAVE;

    // C/D accumulators for the three reductions: sum(mask), sum(prob*mask), sum(prob)
    v8f acc_m = {0.f,0.f,0.f,0.f,0.f,0.f,0.f,0.f};
    v8f acc_p = {0.f,0.f,0.f,0.f,0.f,0.f,0.f,0.f};
    v8f acc_o = {0.f,0.f,0.f,0.f,0.f,0.f,0.f,0.f};
    const v2f ones = {1.0f, 1.0f};

#pragma unroll
    for (int i = 0; i < ITERS; ++i) {
        const int off = base + i * 128 + lane * 4;
        // 16B/lane per tensor: coalesced global_load_b128, non-temporal (streamed once)
        v4f o = __builtin_nontemporal_load((const v4f*)(prob + off));
        v4f m = __builtin_nontemporal_load((const v4f*)(mask + off));

        // Pair adjacent register halves so pk_add/pk_fma hit v[0:1]/v[2:3]
        // directly (no movs). Any bijection onto the 16x4 A-tile is valid
        // because the WMMA-with-ones reduction is permutation-invariant.
        v2f am = m.lo + m.hi;                 // v_pk_add_f32
        v2f ao = o.lo + o.hi;                 // v_pk_add_f32
        v2f ap = o.lo * m.lo + o.hi * m.hi;   // v_pk_mul_f32 + v_pk_fma_f32

        // D = A x ones(4x16) + C : accumulates row-sums of A into every column of C.
        acc_m = __builtin_amdgcn_wmma_f32_16x16x4_f32(false, am, false, ones,
                                                      (short)0, acc_m, false, false);
        acc_p = __builtin_amdgcn_wmma_f32_16x16x4_f32(false, ap, false, ones,
                                                      (short)0, acc_p, false, false);
        acc_o = __builtin_amdgcn_wmma_f32_16x16x4_f32(false, ao, false, ones,
                                                      (short)0, acc_o, false, false);
    }

    // Every column of C holds the same totals; rows M=0..7 live in lanes 0..15,
    // rows M=8..15 in lanes 16..31. Reduce 8 VGPRs per lane, then fold halves.
    float sm = 0.f, sp = 0.f, so = 0.f;
#pragma unroll
    for (int j = 0; j < 8; ++j) { sm += acc_m[j]; sp += acc_p[j]; so += acc_o[j]; }
    sm += __shfl_xor(sm, 16, 32);
    sp += __shfl_xor(sp, 16, 32);
    so += __shfl_xor(so, 16, 32);

    __shared__ float red[WAVES_PER_BLK][3];
    if (lane == 0) { red[wave][0] = sm; red[wave][1] = sp; red[wave][2] = so; }
    __syncthreads();

    if (threadIdx.x == 0) {
        float tm = 0.f, tp = 0.f, to = 0.f;
#pragma unroll
        for (int w = 0; w < WAVES_PER_BLK; ++w) {
            tm += red[w][0]; tp += red[w][1]; to += red[w][2];
        }
        atomicAdd(&ws[bc * 3 + 0], tm);   // sum(mask)
        atomicAdd(&ws[bc * 3 + 1], tp);   // sum(prob*mask)  = inter
        atomicAdd(&ws[bc * 3 + 2], to);   // sum(prob)
    }
}

// ---------------- epilogue: Dice formula on 32 (b,c) partials ----------------
__global__ void dice_final(const float* __restrict__ ws, float* __restrict__ out) {
    const int t = threadIdx.x;            // lane t: b = t/4, c = t%4
    float sm = ws[t * 3 + 0];
    float si = ws[t * 3 + 1];
    float so = ws[t * 3 + 2];
    float total = so + sm;                // sum(prob + mask)
    float w     = 1.0f / (sm * sm + EPS); // class weight
    float nom   = w * si;
    float den   = w * total + EPS;

    // reduce over classes (groups of 4 lanes)
    nom += __shfl_xor(nom, 1, 32); nom += __shfl_xor(nom, 2, 32);
    den += __shfl_xor(den, 1, 32); den += __shfl_xor(den, 2, 32);

    float ps = 1.0f - 2.0f * nom / den;   // per-subject, replicated x4 in group

    // sum over the 8 groups -> sum_b(per_subject) in every lane
    ps += __shfl_xor(ps, 4, 32);
    ps += __shfl_xor(ps, 8, 32);
    ps += __shfl_xor(ps, 16, 32);

    if (t == 0) out[0] = ps * (1.0f / 32.0f);  // sum_b(per_subject) / 32
}

extern "C" void kernel_launch(void* const* d_in, const int* in_sizes, int n_in,
                              void* d_out, int out_size, void* d_ws, size_t ws_size,
                              hipStream_t stream) {
    (void)in_sizes; (void)n_in; (void)out_size; (void)ws_size;
    const float* prob = (const float*)d_in[0];
    const float* mask = (const float*)d_in[1];
    float* ws = (float*)d_ws;

    dice_zero_ws<<<1, 128, 0, stream>>>(ws);
    dice_partials<<<BC * BLOCKS_PER_BC, THREADS, 0, stream>>>(prob, mask, ws);
    dice_final<<<1, 32, 0, stream>>>(ws, (float*)d_out);
}